// GroupedQueryAttentionSequentialBatched_89687507075263
// MI455X (gfx1250) — compile-verified
//
#include <hip/hip_runtime.h>
#include <hip/hip_bf16.h>
#include <math.h>

typedef __attribute__((ext_vector_type(2))) float v2f;
typedef __attribute__((ext_vector_type(4))) float v4f;
typedef __attribute__((ext_vector_type(8))) float v8f;

#define BB 16
#define HH 32
#define GG 8
#define DD 4096
#define KK 128
#define VV 128
#define MM 8192
#define MT 8193   /* M+1 rows in Kc/Vc */

static __device__ __forceinline__ v8f wmma_f32x4(v2f a, v2f b, v8f c) {
    // V_WMMA_F32_16X16X4_F32 : D(16x16 f32) = A(16x4 f32) * B(4x16 f32) + C
    return __builtin_amdgcn_wmma_f32_16x16x4_f32(
        /*neg_a=*/false, a, /*neg_b=*/false, b,
        /*c_mod=*/(short)0, c, /*reuse_a=*/false, /*reuse_b=*/false);
}

// LDS byte offset of a generic pointer that points into __shared__ storage.
static __device__ __forceinline__ unsigned lds_off(const void* p) {
    return (unsigned)(size_t)(__attribute__((address_space(3))) const char*)p;
}

// Issue 16 async row copies (one 16x128-f32 K tile, row-clamped to M) into LDS.
// Each instruction: 32 lanes x 16B = one contiguous 512B row.  ASYNCcnt += 16.
static __device__ __forceinline__ void issue_ktile(const float* __restrict__ Kbg,
                                                   int m0, const float* ldsbase,
                                                   int lane) {
    unsigned lbase = lds_off(ldsbase) + (unsigned)lane * 16u;
#pragma unroll
    for (int i = 0; i < 16; ++i) {
        int rowg = m0 + i;
        if (rowg > MM) rowg = MM;  // clamp: keeps issue count fixed, no OOB
        const float* gp = Kbg + (size_t)rowg * KK + lane * 4;
        unsigned loff = lbase + (unsigned)(i * 512);
        asm volatile("global_load_async_to_lds_b128 %0, %1, off"
                     :: "v"(loff), "v"(gp)
                     : "memory");
    }
}

// ---------------------------------------------------------------------------
// Kernel 1: stream prev_K -> Kc[:, :, 0:M, :] (and prev_V -> Vc) with NT hints.
// ---------------------------------------------------------------------------
__global__ __launch_bounds__(256) void kv_copy_kernel(const float* __restrict__ src,
                                                      float* __restrict__ dst) {
    const unsigned long nvec_bg = (unsigned long)MM * KK / 4;  // 2^18
    const unsigned long dvec_bg = (unsigned long)MT * KK / 4;  // 262176
    const unsigned long ntot = (unsigned long)BB * GG * nvec_bg;
    const v4f* s = (const v4f*)src;
    v4f* d = (v4f*)dst;
    unsigned long stride = (unsigned long)gridDim.x * blockDim.x;
    for (unsigned long i = (unsigned long)blockIdx.x * blockDim.x + threadIdx.x;
         i < ntot; i += stride) {
        unsigned long bg  = i >> 18;
        unsigned long off = i & (nvec_bg - 1);
        v4f val = __builtin_nontemporal_load(&s[i]);
        __builtin_nontemporal_store(val, &d[bg * dvec_bg + off]);
    }
}

// ---------------------------------------------------------------------------
// Kernel 2: projections q = x*Wq, k_new = x*Wk, v_new = x*Wv with f32 WMMA.
// ---------------------------------------------------------------------------
__global__ __launch_bounds__(256) void proj_kernel(const float* __restrict__ x,
                                                   const float* __restrict__ Wq,
                                                   const float* __restrict__ Wk,
                                                   const float* __restrict__ Wv,
                                                   float* __restrict__ q_ws,
                                                   float* __restrict__ Kc,
                                                   float* __restrict__ Vc) {
    int wave = threadIdx.x >> 5;
    int lane = threadIdx.x & 31;
    int tile = blockIdx.x * 8 + wave;
    if (tile >= 384) return;

    const float* Wslab;
    int slab, kk0, mode;
    if (tile < 256)      { mode = 0; slab = tile >> 3;         kk0 = (tile & 7) * 16;         Wslab = Wq + (size_t)slab * DD * KK; }
    else if (tile < 320) { mode = 1; slab = (tile - 256) >> 3; kk0 = ((tile - 256) & 7) * 16; Wslab = Wk + (size_t)slab * DD * KK; }
    else                 { mode = 2; slab = (tile - 320) >> 3; kk0 = ((tile - 320) & 7) * 16; Wslab = Wv + (size_t)slab * DD * KK; }

    int mrow  = lane & 15;
    int khalf = (lane >> 4) * 2;
    int n     = lane & 15;

    v8f acc = {0.f,0.f,0.f,0.f,0.f,0.f,0.f,0.f};
    for (int d0 = 0; d0 < DD; d0 += 4) {
        int ds = d0 + khalf;
        v2f a;  a.x  = x[(size_t)mrow * DD + ds];
                a.y  = x[(size_t)mrow * DD + ds + 1];
        v2f bv; bv.x = Wslab[(size_t)ds       * KK + kk0 + n];
                bv.y = Wslab[(size_t)(ds + 1) * KK + kk0 + n];
        acc = wmma_f32x4(a, bv, acc);
    }
    int rbase = (lane >> 4) << 3;
    for (int j = 0; j < 8; ++j) {
        int row = j + rbase;  // batch
        float val = acc[j];
        if (mode == 0) {
            q_ws[(size_t)row * HH * KK + (size_t)slab * KK + kk0 + n] = val;
        } else if (mode == 1) {
            Kc[(((size_t)row * GG + slab) * MT + MM) * KK + kk0 + n] = val;
        } else {
            Vc[(((size_t)row * GG + slab) * MT + MM) * KK + kk0 + n] = val;
        }
    }
}

// ---------------------------------------------------------------------------
// Kernel 3: attention. One workgroup per (b,g); 8 waves stripe the 513 m-tiles.
// K tiles streamed by the async engine into double-buffered LDS; WMMA reads
// A and B fragments from LDS (ds_load_b64); online softmax per wave; merge in LDS.
// ---------------------------------------------------------------------------
__global__ __launch_bounds__(256) void attn_kernel(const float* __restrict__ q_ws,
                                                   const float* __restrict__ Kc,
                                                   const float* __restrict__ Vc,
                                                   float* __restrict__ o_ws) {
    __shared__ float ktile[8][2][16 * KK];  // 128 KB: per-wave double-buffered K tiles
    __shared__ float qlds[4][KK];           // 2 KB: group's 4 heads of q
    __shared__ float sc[8][16][4];          // per-wave score staging (m x head)
    __shared__ float red_o[8][4][128];      // per-wave o partials
    __shared__ float red_m[8][4];
    __shared__ float red_s[8][4];

    int bg = blockIdx.x;
    int b = bg >> 3, g = bg & 7;
    const float* Kbg = Kc + (size_t)bg * MT * KK;
    const float* Vbg = Vc + (size_t)bg * MT * KK;

    int wave = threadIdx.x >> 5;
    int lane = threadIdx.x & 31;
    int mloc  = lane & 15;
    int khalf = (lane >> 4) * 2;
    int n     = lane & 15;

    // stage q for the 4 heads {g, g+8, g+16, g+24} into LDS (once per block)
    for (int idx = threadIdx.x; idx < 4 * KK; idx += 256) {
        int ri = idx >> 7, k = idx & 127;
        qlds[ri][k] = q_ws[((size_t)b * HH + (size_t)ri * GG + g) * KK + k];
    }
    __syncthreads();

    const float* qrow = &qlds[n & 3][0];  // B-matrix column n -> head (n&3)

    float mx[4]   = {-INFINITY, -INFINITY, -INFINITY, -INFINITY};
    float ssum[4] = {0.f, 0.f, 0.f, 0.f};
    v4f oacc[4];
    for (int h = 0; h < 4; ++h) oacc[h] = (v4f){0.f, 0.f, 0.f, 0.f};

    const int NT = (MT + 15) / 16;  // 513
    int buf = 0;

    // prologue: async-stage this wave's first tile (ASYNCcnt = 16)
    issue_ktile(Kbg, wave * 16, &ktile[wave][0][0], lane);

    for (int t = wave; t < NT; t += 8) {
        int m0 = t * 16;
        bool has_next = (t + 8) < NT;
        if (has_next) {
            // prior ds reads of the other buffer must retire before overwrite
            asm volatile("s_wait_dscnt 0x0" ::: "memory");
            issue_ktile(Kbg, (t + 8) * 16, &ktile[wave][buf ^ 1][0], lane);
            // async loads complete in order: <=16 outstanding => current tile done
            asm volatile("s_wait_asynccnt 0x10" ::: "memory");
        } else {
            asm volatile("s_wait_asynccnt 0x0" ::: "memory");
        }

        // prefetch this wave's next V tile
        {
            int pm = m0 + 128; if (pm > MM) pm = MM;
            __builtin_prefetch(Vbg + (size_t)pm * KK + lane * 16, 0, 1);
        }

        const float* kt = &ktile[wave][buf][0];
        v8f acc = {0.f,0.f,0.f,0.f,0.f,0.f,0.f,0.f};
#pragma unroll
        for (int kc = 0; kc < 32; ++kc) {
            int ks = kc * 4 + khalf;
            v2f a  = *(const v2f*)(kt + mloc * KK + ks);   // ds_load_b64
            v2f bv = *(const v2f*)(qrow + ks);             // ds_load_b64
            acc = wmma_f32x4(a, bv, acc);
        }

        // stage valid score columns (heads 0..3) to LDS
        if (n < 4) {
            int base = (lane >> 4) << 3;
            for (int j = 0; j < 8; ++j) sc[wave][base + j][n] = acc[j];
        }
        __builtin_amdgcn_wave_barrier();  // wave-local ordering; DS in-order per wave

        // tile max -> merge into running max, rescale partials
        float tmax[4] = {-INFINITY, -INFINITY, -INFINITY, -INFINITY};
        for (int m = 0; m < 16; ++m) {
            bool valid = (m0 + m) < MT;
            for (int h = 0; h < 4; ++h) {
                float v = valid ? sc[wave][m][h] : -INFINITY;
                tmax[h] = fmaxf(tmax[h], v);
            }
        }
        for (int h = 0; h < 4; ++h) {
            float nm = fmaxf(mx[h], tmax[h]);
            float scale = __expf(mx[h] - nm);
            ssum[h] *= scale;
            oacc[h] *= scale;
            mx[h] = nm;
        }
        // accumulate o += w * V rows (lane owns v-columns lane*4 .. lane*4+3)
        for (int m = 0; m < 16; ++m) {
            int mr2 = m0 + m;
            if (mr2 >= MT) break;
            v4f vrow = *(const v4f*)(Vbg + (size_t)mr2 * KK + lane * 4);
            for (int h = 0; h < 4; ++h) {
                float w = __expf(sc[wave][m][h] - mx[h]);
                ssum[h] += w;
                oacc[h] += w * vrow;
            }
        }
        buf ^= 1;
    }

    // publish per-wave partials
    if (lane == 0) {
        for (int h = 0; h < 4; ++h) { red_m[wave][h] = mx[h]; red_s[wave][h] = ssum[h]; }
    }
    for (int h = 0; h < 4; ++h) {
        *(v4f*)&red_o[wave][h][lane * 4] = oacc[h];
    }
    __syncthreads();

    // cross-wave merge: 4 heads x 128 v = 512 outputs over 256 threads
    for (int idx = threadIdx.x; idx < 4 * 128; idx += 256) {
        int h = idx >> 7, v = idx & 127;
        float gm = -INFINITY;
        for (int w = 0; w < 8; ++w) gm = fmaxf(gm, red_m[w][h]);
        float st = 0.f, ov = 0.f;
        for (int w = 0; w < 8; ++w) {
            float e = __expf(red_m[w][h] - gm);
            st += red_s[w][h] * e;
            ov += red_o[w][h][v] * e;
        }
        o_ws[((size_t)b * HH + (size_t)h * GG + g) * VV + v] = ov / st;
    }
}

// ---------------------------------------------------------------------------
// Kernel 4: y[16 x 4096] = o[16 x (H*V)] * Wo[(H*V) x 4096] with f32 WMMA.
// ---------------------------------------------------------------------------
__global__ __launch_bounds__(256) void outproj_kernel(const float* __restrict__ o_ws,
                                                      const float* __restrict__ Wo,
                                                      float* __restrict__ y) {
    int wave = threadIdx.x >> 5;
    int lane = threadIdx.x & 31;
    int tile = blockIdx.x * 8 + wave;   // 256 column tiles
    if (tile >= 256) return;
    int c0 = tile * 16;
    int mrow  = lane & 15;
    int khalf = (lane >> 4) * 2;
    int n     = lane & 15;

    v8f acc = {0.f,0.f,0.f,0.f,0.f,0.f,0.f,0.f};
    for (int k0 = 0; k0 < HH * VV; k0 += 4) {
        int ks = k0 + khalf;
        v2f a;  a.x  = o_ws[(size_t)mrow * HH * VV + ks];
                a.y  = o_ws[(size_t)mrow * HH * VV + ks + 1];
        v2f bv; bv.x = Wo[(size_t)ks       * DD + c0 + n];
                bv.y = Wo[(size_t)(ks + 1) * DD + c0 + n];
        acc = wmma_f32x4(a, bv, acc);
    }
    int rbase = (lane >> 4) << 3;
    for (int j = 0; j < 8; ++j) {
        int row = j + rbase;
        y[(size_t)row * DD + c0 + n] = acc[j];
    }
}

// ---------------------------------------------------------------------------
extern "C" void kernel_launch(void* const* d_in, const int* in_sizes, int n_in,
                              void* d_out, int out_size, void* d_ws, size_t ws_size,
                              hipStream_t stream) {
    (void)in_sizes; (void)n_in; (void)out_size; (void)ws_size;
    const float* x      = (const float*)d_in[0];
    const float* prev_K = (const float*)d_in[1];
    const float* prev_V = (const float*)d_in[2];
    const float* Wq     = (const float*)d_in[3];
    const float* Wk     = (const float*)d_in[4];
    const float* Wv     = (const float*)d_in[5];
    const float* Wo     = (const float*)d_in[6];

    float* y  = (float*)d_out;
    float* Kc = y + (size_t)BB * DD;                       // [B,G,M+1,K]
    float* Vc = Kc + (size_t)BB * GG * MT * KK;            // [B,G,M+1,V]

    float* q_ws = (float*)d_ws;                            // [B,H,K]
    float* o_ws = q_ws + (size_t)BB * HH * KK;             // [B,H,V]

    hipLaunchKernelGGL(kv_copy_kernel, dim3(4096), dim3(256), 0, stream, prev_K, Kc);
    hipLaunchKernelGGL(kv_copy_kernel, dim3(4096), dim3(256), 0, stream, prev_V, Vc);
    hipLaunchKernelGGL(proj_kernel,    dim3(48),   dim3(256), 0, stream,
                       x, Wq, Wk, Wv, q_ws, Kc, Vc);
    hipLaunchKernelGGL(attn_kernel,    dim3(BB * GG), dim3(256), 0, stream,
                       q_ws, Kc, Vc, o_ws);
    hipLaunchKernelGGL(outproj_kernel, dim3(32),   dim3(256), 0, stream,
                       o_ws, Wo, y);
}